// InteractionBlock_28819230556709
// MI455X (gfx1250) — compile-verified
//
#include <hip/hip_runtime.h>
#include <math.h>

// ---------------------------------------------------------------------------
// InteractionBlock for MI455X (gfx1250), fp32 end-to-end via V_WMMA_F32_16X16X4_F32
// N=16000, E=512000, F=128, NRBF=64, H=8, NZ=16
// ---------------------------------------------------------------------------

typedef __attribute__((ext_vector_type(2))) float v2f;
typedef __attribute__((ext_vector_type(8))) float v8f;

#define FDIM 128
#define NRBF 64
#define NZD  16
#define CUT  5.0f
#define PI_OVER_CUT 0.6283185307179586f

// D = A(16x4) * B(4x16) + C(16x16), all f32, wave32.
// A/B lane layout: value at [M or N = lane%16][K = 2*(lane/16) + vgpr]
// C/D lane layout: vgpr r holds [M = r + 8*(lane/16)][N = lane%16]
__device__ __forceinline__ v8f wmma4(v2f a, v2f b, v8f c) {
  return __builtin_amdgcn_wmma_f32_16x16x4_f32(
      /*neg_a=*/false, a, /*neg_b=*/false, b,
      /*c_mod=*/(short)0, c, /*reuse_a=*/false, /*reuse_b=*/false);
}

// fast silu: v_exp_f32 + v_add + v_rcp_f32 + v_mul (no IEEE div chain)
__device__ __forceinline__ float silu(float x) {
  return x * __builtin_amdgcn_rcpf(1.0f + __expf(-x));
}

// ---------------------------------------------------------------------------
// Zero workspace (y + cnt)
// ---------------------------------------------------------------------------
__global__ void k_zero(float* __restrict__ p, long long n) {
  long long i = (long long)blockIdx.x * blockDim.x + threadIdx.x;
  long long stride = (long long)gridDim.x * blockDim.x;
  for (; i < n; i += stride) p[i] = 0.0f;
}

// ---------------------------------------------------------------------------
// out[r0..r0+15, :] = (silu?)(X) @ W^T + bias     (one wave per 16-row tile)
// W is [128,128] row-major; out[n,f] = sum_k X[n,k]*W[f,k]
// ---------------------------------------------------------------------------
template <bool SILU_IN, bool HAS_BIAS>
__global__ __launch_bounds__(256) void k_lin(const float* __restrict__ X,
                                             const float* __restrict__ W,
                                             const float* __restrict__ bias,
                                             float* __restrict__ out,
                                             int Nn) {
  int wv = threadIdx.x >> 5, lane = threadIdx.x & 31;
  int l16 = lane & 15, half = lane >> 4;
  int r0 = (blockIdx.x * 8 + wv) * 16;
  if (r0 >= Nn) return;
  v8f acc[8] = {};
  for (int k = 0; k < FDIM; k += 4) {
    v2f a;
    float a0 = X[(size_t)(r0 + l16) * FDIM + k + 2 * half];
    float a1 = X[(size_t)(r0 + l16) * FDIM + k + 2 * half + 1];
    if (SILU_IN) { a0 = silu(a0); a1 = silu(a1); }
    a.x = a0; a.y = a1;
#pragma unroll
    for (int t = 0; t < 8; t++) {
      const float* br = W + (size_t)(t * 16 + l16) * FDIM + k + 2 * half;
      v2f b; b.x = br[0]; b.y = br[1];
      acc[t] = wmma4(a, b, acc[t]);
    }
  }
#pragma unroll
  for (int t = 0; t < 8; t++) {
    float bb = HAS_BIAS ? bias[t * 16 + l16] : 0.0f;
#pragma unroll
    for (int r = 0; r < 8; r++)
      out[(size_t)(r0 + r + 8 * half) * FDIM + t * 16 + l16] = acc[t][r] + bb;
  }
}

// ---------------------------------------------------------------------------
// Edge kernel: W = (silu(attr@fw1^T+fb1)@fw2^T+fb2)*C ; m = xl[src]*W ;
// atomic scatter-add into y[dst*NZ + z[src]] and cnt. One wave per 16 edges.
// ---------------------------------------------------------------------------
__global__ __launch_bounds__(128) void k_edges(const float* __restrict__ edge_attr,
                                               const int* __restrict__ edge_index,
                                               const int* __restrict__ z,
                                               const float* __restrict__ edge_weight,
                                               const float* __restrict__ fw1,
                                               const float* __restrict__ fb1,
                                               const float* __restrict__ fw2,
                                               const float* __restrict__ fb2,
                                               const float* __restrict__ xl,
                                               float* __restrict__ y,
                                               float* __restrict__ cnt, int E) {
  __shared__ float h1s[4][NZD * FDIM];  // silu(h1) staging, per wave
  int wv = threadIdx.x >> 5, lane = threadIdx.x & 31;
  int l16 = lane & 15, half = lane >> 4;
  int e0 = (blockIdx.x * 4 + wv) * 16;
  if (e0 >= E) return;
  float* h1 = h1s[wv];
  const int* srcA = edge_index;      // row 0 = src
  const int* dstA = edge_index + E;  // row 1 = dst

  // occupancy counter: one atomic per edge
  if (lane < 16) {
    int e = e0 + lane;
    atomicAdd(&cnt[(size_t)dstA[e] * NZD + z[srcA[e]]], 1.0f);
  }

  // GEMM1: [16,64] @ fw1^T -> [16,128], +bias, silu, stage to LDS
  v8f acc[8] = {};
  for (int k = 0; k < NRBF; k += 4) {
    const float* ar = edge_attr + (size_t)(e0 + l16) * NRBF + k + 2 * half;
    v2f a; a.x = ar[0]; a.y = ar[1];
#pragma unroll
    for (int t = 0; t < 8; t++) {
      const float* br = fw1 + (size_t)(t * 16 + l16) * NRBF + k + 2 * half;
      v2f b; b.x = br[0]; b.y = br[1];
      acc[t] = wmma4(a, b, acc[t]);
    }
  }
#pragma unroll
  for (int t = 0; t < 8; t++) {
    float bias = fb1[t * 16 + l16];
#pragma unroll
    for (int r = 0; r < 8; r++)
      h1[(r + 8 * half) * FDIM + t * 16 + l16] = silu(acc[t][r] + bias);
  }

  // GEMM2: [16,128] @ fw2^T -> [16,128]  (same-wave LDS ops are in-order)
  v8f acc2[8] = {};
  for (int k = 0; k < FDIM; k += 4) {
    v2f a;
    a.x = h1[l16 * FDIM + k + 2 * half];
    a.y = h1[l16 * FDIM + k + 2 * half + 1];
#pragma unroll
    for (int t = 0; t < 8; t++) {
      const float* br = fw2 + (size_t)(t * 16 + l16) * FDIM + k + 2 * half;
      v2f b; b.x = br[0]; b.y = br[1];
      acc2[t] = wmma4(a, b, acc2[t]);
    }
  }

  // epilogue: +fb2, * cutoff, * xl[src], scatter-add
#pragma unroll
  for (int r = 0; r < 8; r++) {
    int e = e0 + r + 8 * half;
    int s = srcA[e], d = dstA[e];
    float wgt = edge_weight[e];
    float Ccut = (wgt < CUT) ? 0.5f * (__cosf(wgt * PI_OVER_CUT) + 1.0f) : 0.0f;
    float* yrow = y + ((size_t)d * NZD + z[s]) * FDIM;
    const float* xlrow = xl + (size_t)s * FDIM;
#pragma unroll
    for (int t = 0; t < 8; t++) {
      int col = t * 16 + l16;
      float Wv = acc2[t][r] + fb2[col];
      atomicAdd(&yrow[col], Wv * Ccut * xlrow[col]);
    }
  }
}

// ---------------------------------------------------------------------------
// Per-node attention: QKV proj (WMMA) -> 8 heads of 16x16 attention (WMMA)
// -> output proj (WMMA) -> masked row-sum. One wave per node (NZ=16 rows).
// ---------------------------------------------------------------------------
__global__ __launch_bounds__(64) void k_attn(const float* __restrict__ y,
                                             const float* __restrict__ cnt,
                                             const float* __restrict__ qw, const float* __restrict__ qb,
                                             const float* __restrict__ kw, const float* __restrict__ kb,
                                             const float* __restrict__ vw, const float* __restrict__ vb,
                                             const float* __restrict__ ow, const float* __restrict__ ob,
                                             float* __restrict__ nodeBuf, int Nn) {
  __shared__ float sqs[2][NZD * FDIM];
  __shared__ float sks[2][NZD * FDIM];
  __shared__ float svs[2][NZD * FDIM];
  __shared__ float sas[2][NZD * NZD];
  int wv = threadIdx.x >> 5, lane = threadIdx.x & 31;
  int l16 = lane & 15, half = lane >> 4;
  int node = blockIdx.x * 2 + wv;
  if (node >= Nn) return;
  float* sq = sqs[wv]; float* sk = sks[wv]; float* sv = svs[wv]; float* sa = sas[wv];
  const float* yg = y + (size_t)node * (NZD * FDIM);

  bool maskj = cnt[(size_t)node * NZD + l16] > 0.0f;  // source-slot mask (j = lane%16)
  bool maski[8];
#pragma unroll
  for (int r = 0; r < 8; r++)
    maski[r] = cnt[(size_t)node * NZD + r + 8 * half] > 0.0f;  // row mask (i = r+8*half)

  // ---- q,k,v = y @ {qw,kw,vw}^T + bias, staged to LDS ----
  const float* Wp[3] = {qw, kw, vw};
  const float* Bp[3] = {qb, kb, vb};
  float* Sp[3] = {sq, sk, sv};
  for (int p = 0; p < 3; p++) {
    v8f acc[8] = {};
    const float* W = Wp[p];
    for (int k = 0; k < FDIM; k += 4) {
      v2f a;
      a.x = yg[l16 * FDIM + k + 2 * half];
      a.y = yg[l16 * FDIM + k + 2 * half + 1];
#pragma unroll
      for (int t = 0; t < 8; t++) {
        const float* br = W + (size_t)(t * 16 + l16) * FDIM + k + 2 * half;
        v2f b; b.x = br[0]; b.y = br[1];
        acc[t] = wmma4(a, b, acc[t]);
      }
    }
#pragma unroll
    for (int t = 0; t < 8; t++) {
      float bias = Bp[p][t * 16 + l16];
#pragma unroll
      for (int r = 0; r < 8; r++)
        Sp[p][(r + 8 * half) * FDIM + t * 16 + l16] = acc[t][r] + bias;
    }
  }

  // ---- per-head 16x16 attention; o_h overwrites q_h's columns in sq ----
  for (int h = 0; h < 8; h++) {
    v8f lg = {};
#pragma unroll
    for (int k = 0; k < 16; k += 4) {
      v2f a, b;
      a.x = sq[l16 * FDIM + h * 16 + k + 2 * half];
      a.y = sq[l16 * FDIM + h * 16 + k + 2 * half + 1];
      b.x = sk[l16 * FDIM + h * 16 + k + 2 * half];      // B[kk][j] = k[j][h*16+kk]
      b.y = sk[l16 * FDIM + h * 16 + k + 2 * half + 1];
      lg = wmma4(a, b, lg);
    }
    // mask + softmax over j (16 lanes within each half-wave), stage attn to LDS
#pragma unroll
    for (int r = 0; r < 8; r++) {
      float v = maskj ? lg[r] : -1e9f;
      float m = v;
      m = fmaxf(m, __shfl_xor(m, 1, 32));
      m = fmaxf(m, __shfl_xor(m, 2, 32));
      m = fmaxf(m, __shfl_xor(m, 4, 32));
      m = fmaxf(m, __shfl_xor(m, 8, 32));
      float e = __expf(v - m);
      float s = e;
      s += __shfl_xor(s, 1, 32);
      s += __shfl_xor(s, 2, 32);
      s += __shfl_xor(s, 4, 32);
      s += __shfl_xor(s, 8, 32);
      sa[(r + 8 * half) * NZD + l16] = e * __builtin_amdgcn_rcpf(s);  // s >= 1
    }
    // o_h = attn @ v_h
    v8f oc = {};
#pragma unroll
    for (int k = 0; k < 16; k += 4) {
      v2f a, b;
      a.x = sa[l16 * NZD + k + 2 * half];
      a.y = sa[l16 * NZD + k + 2 * half + 1];
      b.x = sv[(k + 2 * half) * FDIM + h * 16 + l16];    // B[j][d] = v[j][h*16+d]
      b.y = sv[(k + 2 * half + 1) * FDIM + h * 16 + l16];
      oc = wmma4(a, b, oc);
    }
#pragma unroll
    for (int r = 0; r < 8; r++)
      sq[(r + 8 * half) * FDIM + h * 16 + l16] = oc[r];  // q_h dead -> reuse
  }

  // ---- output projection + masked sum over group rows ----
  v8f accP[8] = {};
  for (int k = 0; k < FDIM; k += 4) {
    v2f a;
    a.x = sq[l16 * FDIM + k + 2 * half];
    a.y = sq[l16 * FDIM + k + 2 * half + 1];
#pragma unroll
    for (int t = 0; t < 8; t++) {
      const float* br = ow + (size_t)(t * 16 + l16) * FDIM + k + 2 * half;
      v2f b; b.x = br[0]; b.y = br[1];
      accP[t] = wmma4(a, b, accP[t]);
    }
  }
#pragma unroll
  for (int t = 0; t < 8; t++) {
    float bias = ob[t * 16 + l16];
    float s = 0.0f;
#pragma unroll
    for (int r = 0; r < 8; r++) {
      float v = accP[t][r] + bias;
      s += maski[r] ? v : 0.0f;
    }
    s += __shfl_xor(s, 16, 32);  // combine i=0..7 half with i=8..15 half (same column)
    if (half == 0)
      nodeBuf[(size_t)node * FDIM + t * 16 + l16] = s;
  }
}

// ---------------------------------------------------------------------------
extern "C" void kernel_launch(void* const* d_in, const int* in_sizes, int n_in,
                              void* d_out, int out_size, void* d_ws, size_t ws_size,
                              hipStream_t stream) {
  const float* x           = (const float*)d_in[0];
  const int*   z           = (const int*)d_in[1];
  const int*   edge_index  = (const int*)d_in[2];
  const float* edge_weight = (const float*)d_in[3];
  const float* edge_attr   = (const float*)d_in[4];
  const float* lin1_w      = (const float*)d_in[5];
  const float* fw1 = (const float*)d_in[6];
  const float* fb1 = (const float*)d_in[7];
  const float* fw2 = (const float*)d_in[8];
  const float* fb2 = (const float*)d_in[9];
  const float* qw  = (const float*)d_in[10];
  const float* qb  = (const float*)d_in[11];
  const float* kw  = (const float*)d_in[12];
  const float* kb  = (const float*)d_in[13];
  const float* vw  = (const float*)d_in[14];
  const float* vb  = (const float*)d_in[15];
  const float* ow  = (const float*)d_in[16];
  const float* ob  = (const float*)d_in[17];
  const float* lw  = (const float*)d_in[18];
  const float* lb  = (const float*)d_in[19];

  int N = in_sizes[1];  // z has N entries
  int E = in_sizes[3];  // edge_weight has E entries

  // workspace layout (floats): y[N*NZ*F] | cnt[N*NZ] | xl[N*F] | node[N*F]
  float* y       = (float*)d_ws;
  float* cnt     = y + (size_t)N * NZD * FDIM;
  float* xl      = cnt + (size_t)N * NZD;
  float* nodeBuf = xl + (size_t)N * FDIM;

  long long nzero = (long long)N * NZD * FDIM + (long long)N * NZD;  // y + cnt contiguous
  k_zero<<<2048, 256, 0, stream>>>(y, nzero);

  int rowTiles = N / 16;  // N divisible by 16
  k_lin<false, false><<<(rowTiles + 7) / 8, 256, 0, stream>>>(x, lin1_w, nullptr, xl, N);

  int edgeTiles = E / 16;  // E divisible by 16
  k_edges<<<(edgeTiles + 3) / 4, 128, 0, stream>>>(edge_attr, edge_index, z, edge_weight,
                                                   fw1, fb1, fw2, fb2, xl, y, cnt, E);

  k_attn<<<(N + 1) / 2, 64, 0, stream>>>(y, cnt, qw, qb, kw, kb, vw, vb, ow, ob, nodeBuf, N);

  k_lin<true, true><<<(rowTiles + 7) / 8, 256, 0, stream>>>(nodeBuf, lw, lb, (float*)d_out, N);
}